// GCN_21784074125836
// MI455X (gfx1250) — compile-verified
//
#include <hip/hip_runtime.h>
#include <hip/hip_bf16.h>

typedef __attribute__((ext_vector_type(2))) float v2f;
typedef __attribute__((ext_vector_type(8))) float v8f;

#define F_DIM 64

// ---------------- degree / norm precompute ----------------

__global__ __launch_bounds__(256) void deg_init_kernel(float* __restrict__ deg, int n) {
    int i = blockIdx.x * blockDim.x + threadIdx.x;
    if (i < n) deg[i] = 1.0f;                      // self-loop weight
}

__global__ __launch_bounds__(256) void deg_accum_kernel(const int* __restrict__ col,
                                                        const float* __restrict__ ew,
                                                        float* __restrict__ deg, int e) {
    int i = blockIdx.x * blockDim.x + threadIdx.x;
    if (i < e) atomicAdd(&deg[col[i]], ew[i]);
}

__global__ __launch_bounds__(256) void dinv_kernel(float* __restrict__ deg, int n) {
    int i = blockIdx.x * blockDim.x + threadIdx.x;
    if (i < n) {
        float d = deg[i];                          // d >= 1 always (self-loop)
        deg[i] = (d > 0.0f) ? __frsqrt_rn(d) : 0.0f;
    }
}

__global__ __launch_bounds__(256) void norm_kernel(const int* __restrict__ row,
                                                   const int* __restrict__ col,
                                                   const float* __restrict__ ew,
                                                   const float* __restrict__ dinv,
                                                   float* __restrict__ norm, int e) {
    int i = blockIdx.x * blockDim.x + threadIdx.x;
    if (i < e) norm[i] = dinv[row[i]] * ew[i] * dinv[col[i]];
}

// ---------------- dense GEMM via V_WMMA_F32_16X16X4_F32 ----------------
// H[N,64] = X[N,64] @ W[64,64], fp32 exact.
// One wave computes a 16-row stripe: 4 accumulators (16x16 tiles), 16 k-steps.
// A 16x4 f32 layout: lanes 0-15 -> M=lane, VGPR{0,1} = K{0,1}; lanes 16-31 -> K{2,3}.
// B 4x16 mirrored over N. C/D: VGPR v -> M=v (lanes 0-15) / M=v+8 (lanes 16-31).

__global__ __launch_bounds__(256) void gemm64_wmma_kernel(const float* __restrict__ X,
                                                          const float* __restrict__ W,
                                                          float* __restrict__ H, int n) {
    int wave = (blockIdx.x * blockDim.x + threadIdx.x) >> 5;
    int lane = threadIdx.x & 31;
    int row0 = wave * 16;
    if (row0 >= n) return;                         // uniform per wave; EXEC stays all-1s

    int half = lane >> 4;                          // 0: K pair {0,1}, 1: K pair {2,3}
    int lr   = lane & 15;

    const float* arow = X + (size_t)(row0 + lr) * F_DIM;

    v8f acc0 = {}, acc1 = {}, acc2 = {}, acc3 = {};

    #pragma unroll
    for (int k = 0; k < F_DIM; k += 4) {
        int ka = k + 2 * half;
        v2f a;
        a.x = arow[ka];
        a.y = arow[ka + 1];

        const float* w0 = W + (size_t)ka * F_DIM + lr;        // row ka
        const float* w1 = w0 + F_DIM;                         // row ka+1
        v2f b0, b1, b2, b3;
        b0.x = w0[0];   b0.y = w1[0];
        b1.x = w0[16];  b1.y = w1[16];
        b2.x = w0[32];  b2.y = w1[32];
        b3.x = w0[48];  b3.y = w1[48];

        acc0 = __builtin_amdgcn_wmma_f32_16x16x4_f32(false, a, false, b0, (short)0, acc0, false, false);
        acc1 = __builtin_amdgcn_wmma_f32_16x16x4_f32(false, a, false, b1, (short)0, acc1, false, false);
        acc2 = __builtin_amdgcn_wmma_f32_16x16x4_f32(false, a, false, b2, (short)0, acc2, false, false);
        acc3 = __builtin_amdgcn_wmma_f32_16x16x4_f32(false, a, false, b3, (short)0, acc3, false, false);
    }

    #pragma unroll
    for (int v = 0; v < 8; ++v) {
        int m = row0 + v + 8 * half;
        float* o = H + (size_t)m * F_DIM + lr;
        o[0]  = acc0[v];
        o[16] = acc1[v];
        o[32] = acc2[v];
        o[48] = acc3[v];
    }
}

// ---------------- aggregation ----------------

// agg[i,:] = dinv[i]^2 * h[i,:]   (self-loop contribution; norm_self = 1/deg)
__global__ __launch_bounds__(256) void agg_init_kernel(const float* __restrict__ H,
                                                       const float* __restrict__ dinv,
                                                       float* __restrict__ AGG, int n) {
    int tid = blockIdx.x * blockDim.x + threadIdx.x;   // n*16 threads, float4 each
    int i = tid >> 4;
    if (i >= n) return;
    int f = (tid & 15) * 4;
    float s = dinv[i];
    s = s * s;
    float4 h4 = *(const float4*)(H + (size_t)i * F_DIM + f);
    float4 o;
    o.x = s * h4.x; o.y = s * h4.y; o.z = s * h4.z; o.w = s * h4.w;
    *(float4*)(AGG + (size_t)i * F_DIM + f) = o;
}

// Half-wave (16 lanes) per edge: coalesced 256B gather of h[r,:], scaled
// scatter-add into agg[c,:] (4 x global_atomic_add_f32 per lane).
__global__ __launch_bounds__(256) void scatter_kernel(const int* __restrict__ row,
                                                      const int* __restrict__ col,
                                                      const float* __restrict__ norm,
                                                      const float* __restrict__ H,
                                                      float* __restrict__ AGG, int e) {
    int tid = blockIdx.x * blockDim.x + threadIdx.x;
    int ei = tid >> 4;
    if (ei >= e) return;
    int f = (tid & 15) * 4;
    int r = row[ei];
    int c = col[ei];
    float s = norm[ei];
    float4 h4 = *(const float4*)(H + (size_t)r * F_DIM + f);
    float* dst = AGG + (size_t)c * F_DIM + f;
    atomicAdd(dst + 0, s * h4.x);
    atomicAdd(dst + 1, s * h4.y);
    atomicAdd(dst + 2, s * h4.z);
    atomicAdd(dst + 3, s * h4.w);
}

__global__ __launch_bounds__(256) void bias_relu_kernel(float* __restrict__ Y,
                                                        const float* __restrict__ b,
                                                        int total) {
    int i = blockIdx.x * blockDim.x + threadIdx.x;
    if (i < total) Y[i] = fmaxf(Y[i] + b[i & (F_DIM - 1)], 0.0f);
}

// ---------------- driver ----------------

static inline int cdiv(long long a, long long b) { return (int)((a + b - 1) / b); }

extern "C" void kernel_launch(void* const* d_in, const int* in_sizes, int n_in,
                              void* d_out, int out_size, void* d_ws, size_t ws_size,
                              hipStream_t stream) {
    const float* node_feats = (const float*)d_in[0];
    const int*   edge_index = (const int*)d_in[1];      // [2, E]
    const float* edge_feats = (const float*)d_in[2];
    const float* W1         = (const float*)d_in[3];
    const float* b1         = (const float*)d_in[4];
    const float* W2         = (const float*)d_in[5];
    const float* b2         = (const float*)d_in[6];
    float*       out        = (float*)d_out;

    const int N = in_sizes[0] / F_DIM;                  // 100000
    const int E = in_sizes[1] / 2;                      // 1600000
    const int* row = edge_index;
    const int* col = edge_index + E;

    // workspace carve-up (256B aligned)
    char* ws = (char*)d_ws;
    size_t off = 0;
    auto alloc = [&](size_t bytes) -> char* {
        char* p = ws + off;
        off = (off + bytes + 255) & ~(size_t)255;
        return p;
    };
    float* dinv = (float*)alloc((size_t)N * sizeof(float));
    float* norm = (float*)alloc((size_t)E * sizeof(float));
    float* bufA = (float*)alloc((size_t)N * F_DIM * sizeof(float));   // h
    float* bufB = (float*)alloc((size_t)N * F_DIM * sizeof(float));   // agg / x1
    (void)ws_size;

    const int BT = 256;
    const int gN    = cdiv(N, BT);
    const int gE    = cdiv(E, BT);
    const int gN16  = cdiv((long long)N * 16, BT);
    const int gE16  = cdiv((long long)E * 16, BT);
    const int gNF   = cdiv((long long)N * F_DIM, BT);
    const int gGemm = cdiv((long long)cdiv(N, 16) * 32, BT);  // one wave per 16 rows

    // shared normalization (same for both layers)
    deg_init_kernel <<<gN, BT, 0, stream>>>(dinv, N);
    deg_accum_kernel<<<gE, BT, 0, stream>>>(col, edge_feats, dinv, E);
    dinv_kernel     <<<gN, BT, 0, stream>>>(dinv, N);
    norm_kernel     <<<gE, BT, 0, stream>>>(row, col, edge_feats, dinv, norm, E);

    // layer 1: x1 = relu(agg(norm, x@W1) + b1)
    gemm64_wmma_kernel<<<gGemm, BT, 0, stream>>>(node_feats, W1, bufA, N);
    agg_init_kernel   <<<gN16,  BT, 0, stream>>>(bufA, dinv, bufB, N);
    scatter_kernel    <<<gE16,  BT, 0, stream>>>(row, col, norm, bufA, bufB, E);
    bias_relu_kernel  <<<gNF,   BT, 0, stream>>>(bufB, b1, N * F_DIM);

    // layer 2: out = relu(agg(norm, x1@W2) + b2)
    gemm64_wmma_kernel<<<gGemm, BT, 0, stream>>>(bufB, W2, bufA, N);
    agg_init_kernel   <<<gN16,  BT, 0, stream>>>(bufA, dinv, out, N);
    scatter_kernel    <<<gE16,  BT, 0, stream>>>(row, col, norm, bufA, out, E);
    bias_relu_kernel  <<<gNF,   BT, 0, stream>>>(out, b2, N * F_DIM);
}